// EmbeddingsModule_45311904973549
// MI455X (gfx1250) — compile-verified
//
#include <hip/hip_runtime.h>

#define SEQ  512
#define HID  768
#define NW   256
#define WEMB 300
#define OUTD (HID + WEMB)

typedef __attribute__((ext_vector_type(2))) float v2f;
typedef __attribute__((ext_vector_type(8))) float v8f;

// Kernel 1: starts[b][w] = lower_bound(token_ids[b,:], w); starts[b][NW] = SEQ.
// Segments of word w are [starts[w], starts[w+1]) because token_ids is sorted.
__global__ void __launch_bounds__(256)
seg_bounds_kernel(const int* __restrict__ tok, int* __restrict__ starts) {
    const int b = blockIdx.x;
    const int w = threadIdx.x;
    const int* t = tok + b * SEQ;
    int lo = 0, hi = SEQ;
    while (lo < hi) {
        int mid = (lo + hi) >> 1;
        if (t[mid] < w) lo = mid + 1; else hi = mid;
    }
    starts[b * (NW + 1) + w] = lo;
    if (w == 0) starts[b * (NW + 1) + NW] = SEQ;
}

// Kernel 2: segment-mean as indicator-matmul on the fp32 WMMA pipe.
// Block = (batch b, 16-word tile w0). 8 waves; wave owns 6 H-tiles of 16 cols
// with 6 live accumulators. Per k-step: build A (16x4 0/1 indicator) once,
// then 6 coalesced B load-pairs + 6 independent v_wmma_f32_16x16x4_f32.
// Layout per ISA 7.12.2 (32-bit 16x4 A): lane = 16*kh + m; A vgpr0 -> K=2*kh,
// vgpr1 -> K=2*kh+1, row M = m.  C/D: vgpr r -> row M = r + 8*kh, col N = m.
__global__ void __launch_bounds__(256)
seg_mean_wmma_kernel(const float* __restrict__ hidden,
                     const int* __restrict__ tok,
                     const int* __restrict__ starts,
                     float* __restrict__ out) {
    const int b    = blockIdx.x >> 4;
    const int w0   = (blockIdx.x & 15) << 4;
    const int lane = threadIdx.x & 31;
    const int wave = threadIdx.x >> 5;
    const int m    = lane & 15;
    const int kh   = lane >> 4;
    const int wm   = w0 + m;              // this lane's word id (A rows)

    const int* st = starts + b * (NW + 1) + w0;
    const int s0 = st[0];                 // uniform across wave
    const int s1 = st[16];                // uniform across wave
    const int* t = tok + b * SEQ;
    const float* hb = hidden + (size_t)b * SEQ * HID;

    // Reciprocal token counts for this lane's 8 output rows (M = r + 8*kh):
    // 8 divides here instead of 48 in the epilogue.
    float inv[8];
#pragma unroll
    for (int r = 0; r < 8; ++r) {
        const int M = r + 8 * kh;
        const int c = st[M + 1] - st[M];
        inv[r] = 1.0f / (float)(c > 0 ? c : 1);
    }

    v8f acc[6];
#pragma unroll
    for (int i = 0; i < 6; ++i) acc[i] = {};

    // Induction-variable pointers: this lane covers rows kbase, kbase+1 per step.
    const int kbase = s0 + 2 * kh;
    const int* tp  = t + kbase;
    const float* p = hb + (size_t)kbase * HID + m;

    int k = s0;
    // Main body: k+4 <= s1  =>  all lane rows < s1, no clamping, no branches.
    for (; k + 4 <= s1; k += 4, tp += 4, p += 4 * HID) {
        const int tk0 = tp[0];
        const int tk1 = tp[1];
        v2f a;
        a.x = (tk0 == wm) ? 1.0f : 0.0f;
        a.y = (tk1 == wm) ? 1.0f : 0.0f;

        // gfx1250 speculative prefetch of next k-step's rows.
        __builtin_prefetch(p + 4 * HID + (wave << 4), 0, 1);

#pragma unroll
        for (int i = 0; i < 6; ++i) {
            const int h0 = (wave + 8 * i) << 4;
            v2f bv;
            bv.x = p[h0];
            bv.y = p[h0 + HID];
            acc[i] = __builtin_amdgcn_wmma_f32_16x16x4_f32(
                false, a, false, bv, (short)0, acc[i], false, false);
        }
    }

    // Branchless tail: 0 < s1 - k < 4. Select-clamp rows; A==0 exactly
    // cancels the clamped B data (0/1 fp32 indicator), loads stay unconditional.
    if (k < s1) {
        const int k0 = k + 2 * kh;
        const int k1 = k0 + 1;
        const bool c0 = k0 < s1;
        const bool c1 = k1 < s1;
        const int r0 = c0 ? k0 : s0;      // any in-range row
        const int r1 = c1 ? k1 : s0;
        const int tk0 = t[r0];
        const int tk1 = t[r1];
        v2f a;
        a.x = (c0 & (tk0 == wm)) ? 1.0f : 0.0f;
        a.y = (c1 & (tk1 == wm)) ? 1.0f : 0.0f;
        const float* p0 = hb + (size_t)r0 * HID + m;
        const float* p1 = hb + (size_t)r1 * HID + m;
#pragma unroll
        for (int i = 0; i < 6; ++i) {
            const int h0 = (wave + 8 * i) << 4;
            v2f bv;
            bv.x = p0[h0];
            bv.y = p1[h0];
            acc[i] = __builtin_amdgcn_wmma_f32_16x16x4_f32(
                false, a, false, bv, (short)0, acc[i], false, false);
        }
    }

    // Scale by reciprocal count and store (lanes 0..15 -> 64B coalesced/row).
#pragma unroll
    for (int i = 0; i < 6; ++i) {
        const int h0 = (wave + 8 * i) << 4;
#pragma unroll
        for (int r = 0; r < 8; ++r) {
            const int M = r + 8 * kh;
            out[((size_t)(b * NW + w0 + M)) * OUTD + h0 + m] = acc[i][r] * inv[r];
        }
    }
}

// Kernel 3: word2vec gather into the last 300 columns, as float4 (B128).
// Row bases are 16B-aligned: w2v row = wid*1200 B, out slot = bw*4272+3072 B.
__global__ void __launch_bounds__(96)
gather_kernel(const float* __restrict__ w2v, const int* __restrict__ wids,
              float* __restrict__ out) {
    const int bw = blockIdx.x;   // b*NW + w
    const int e  = threadIdx.x;  // float4 index, 75 per row
    if (e < WEMB / 4) {
        const int wid = wids[bw];
        const float4* src = (const float4*)(w2v + (size_t)wid * WEMB);
        float4* dst = (float4*)(out + (size_t)bw * OUTD + HID);
        dst[e] = src[e];
    }
}

extern "C" void kernel_launch(void* const* d_in, const int* in_sizes, int n_in,
                              void* d_out, int out_size, void* d_ws, size_t ws_size,
                              hipStream_t stream) {
    const float* hidden = (const float*)d_in[0];   // [B, S, H] fp32
    const float* w2v    = (const float*)d_in[1];   // [VOCAB, WE] fp32
    const int*   tok    = (const int*)d_in[2];     // [B, S] int32
    const int*   wids   = (const int*)d_in[3];     // [B, NW] int32
    float* out = (float*)d_out;                    // [B, NW, OUTD] fp32
    int* starts = (int*)d_ws;                      // [B, NW+1] int32 scratch

    const int B = in_sizes[2] / SEQ;

    seg_bounds_kernel<<<B, NW, 0, stream>>>(tok, starts);
    seg_mean_wmma_kernel<<<B * (NW / 16), 256, 0, stream>>>(hidden, tok, starts, out);
    gather_kernel<<<B * NW, 96, 0, stream>>>(w2v, wids, out);
}